// ProbSparseAttention_85478439125335
// MI455X (gfx1250) — compile-verified
//
#include <hip/hip_runtime.h>
#include <math.h>

#define B_  2
#define H_  8
#define L_  4096
#define D_  64
#define NQ_ 45            // num_queries = min(5*ceil(ln 4096), 4096) = 45

typedef float v2f __attribute__((ext_vector_type(2)));
typedef float v8f __attribute__((ext_vector_type(8)));

// ---------------------------------------------------------------------------
// Stage 1: sparsity[b,h,q] = max_j dot(q, key[kidx[q,j]]) - sum_j(...)/L
// One thread per (b,h,q).
// ---------------------------------------------------------------------------
__global__ void sparsity_kernel(const float* __restrict__ query,
                                const float* __restrict__ key,
                                const int*   __restrict__ key_index,
                                float*       __restrict__ sparsity,
                                int nk) {
  int gid = blockIdx.x * blockDim.x + threadIdx.x;     // 0 .. B*H*L-1
  int bh  = gid >> 12;                                 // / 4096
  int q   = gid & (L_ - 1);
  const float4* qv = reinterpret_cast<const float4*>(query + ((size_t)bh * L_ + q) * D_);
  float4 qr[16];
  #pragma unroll
  for (int i = 0; i < 16; ++i) qr[i] = qv[i];
  float mx = -INFINITY, sm = 0.f;
  for (int j = 0; j < nk; ++j) {
    int kidx = key_index[q * nk + j];
    const float4* kv = reinterpret_cast<const float4*>(key + ((size_t)bh * L_ + kidx) * D_);
    float d = 0.f;
    #pragma unroll
    for (int i = 0; i < 16; ++i) {
      float4 k4 = kv[i];
      d += qr[i].x * k4.x + qr[i].y * k4.y + qr[i].z * k4.z + qr[i].w * k4.w;
    }
    mx = fmaxf(mx, d);
    sm += d;
  }
  sparsity[gid] = mx - sm * (1.0f / (float)L_);
}

// ---------------------------------------------------------------------------
// Stage 2: per (b,h), select top-45 sparsity indices (iterative argmax,
// ties -> smaller index, matching jax.lax.top_k). One block per (b,h).
// ---------------------------------------------------------------------------
__global__ void topk_kernel(const float* __restrict__ sparsity,
                            int*         __restrict__ qidx) {
  __shared__ float vals[L_];
  __shared__ float rv[256];
  __shared__ int   ri[256];
  int bh = blockIdx.x, t = threadIdx.x;
  for (int i = t; i < L_; i += 256) vals[i] = sparsity[(size_t)bh * L_ + i];
  __syncthreads();
  for (int iter = 0; iter < NQ_; ++iter) {
    float bv = -INFINITY; int bi = 0;
    for (int i = t; i < L_; i += 256) {        // increasing i -> keeps lowest tie
      float v = vals[i];
      if (v > bv) { bv = v; bi = i; }
    }
    rv[t] = bv; ri[t] = bi;
    __syncthreads();
    for (int s = 128; s > 0; s >>= 1) {
      if (t < s) {
        float ov = rv[t + s]; int oi = ri[t + s];
        if (ov > rv[t] || (ov == rv[t] && oi < ri[t])) { rv[t] = ov; ri[t] = oi; }
      }
      __syncthreads();
    }
    if (t == 0) {
      qidx[bh * NQ_ + iter] = ri[0];
      vals[ri[0]] = -INFINITY;
    }
    __syncthreads();
  }
}

// ---------------------------------------------------------------------------
// Stage 3: context = cumsum(value, axis=L). One block per (b,h), lane = d.
// Fully coalesced streaming scan.
// ---------------------------------------------------------------------------
__global__ void cumsum_kernel(const float* __restrict__ value,
                              float*       __restrict__ out) {
  int bh = blockIdx.x, d = threadIdx.x;
  const float* vp = value + (size_t)bh * L_ * D_ + d;
  float*       op = out   + (size_t)bh * L_ * D_ + d;
  float acc = 0.f;
  for (int q = 0; q < L_; ++q) {
    acc += vp[(size_t)q * D_];
    op[(size_t)q * D_] = acc;
  }
}

// ---------------------------------------------------------------------------
// Stage 4: dense causal attention for the 45 selected queries, scattered into
// the cumsum output. One wave per (b,h, 16-query tile): 16*3 = 48 blocks.
// Q.K^T tiles via V_WMMA_F32_16X16X4_F32 (fp32, K chained 16x).
// Online (flash) softmax over 256 key-tiles of 16.
// ---------------------------------------------------------------------------
__global__ void __launch_bounds__(32)
attn_kernel(const float* __restrict__ query,
            const float* __restrict__ key,
            const float* __restrict__ value,
            const int*   __restrict__ qidx,
            float*       __restrict__ out) {
  __shared__ float ptile[16 * 16];
  __shared__ float lm[16], ll[16], lscale[16];
  __shared__ int   lqi[16];

  int bh   = blockIdx.x / 3;
  int rt   = blockIdx.x % 3;          // query row-tile (0..2), rows rt*16..rt*16+15 of 48
  int lane = threadIdx.x;             // wave32
  int half = lane >> 4;               // 0: lanes 0-15, 1: lanes 16-31
  int n    = lane & 15;
  int aoff = half * 2;                // K sub-offset within each 4-chunk

  // Row owned by this lane for the A matrix (M = lane & 15).
  int grow_a = rt * 16 + n;
  int qia    = (grow_a < NQ_) ? qidx[bh * NQ_ + grow_a] : 0;
  if (lane < 16) { lqi[lane] = qia; lm[lane] = -INFINITY; ll[lane] = 0.f; lscale[lane] = 0.f; }
  __syncthreads();

  // Preload A (16x64 query tile) in WMMA A-layout: a[k] covers K = 4k..4k+3.
  const float* qp = query + ((size_t)bh * L_ + qia) * D_;
  v2f a[16];
  #pragma unroll
  for (int k = 0; k < 16; ++k) {
    a[k][0] = qp[4 * k + aoff];
    a[k][1] = qp[4 * k + aoff + 1];
  }

  float2 oacc[16];
  #pragma unroll
  for (int r = 0; r < 16; ++r) { oacc[r].x = 0.f; oacc[r].y = 0.f; }

  const float* keyb = key   + (size_t)bh * L_ * D_;
  const float* valb = value + (size_t)bh * L_ * D_;

  for (int kt = 0; kt < L_ / 16; ++kt) {
    // ---- 16x16 score tile: 16 chained fp32 WMMAs (K = 64) ----
    v8f sacc = {};
    const float* kp = keyb + (size_t)(kt * 16 + n) * D_ + aoff;  // B col = lane&15
    #pragma unroll
    for (int k = 0; k < 16; ++k) {
      v2f b;
      b[0] = kp[4 * k];
      b[1] = kp[4 * k + 1];
      sacc = __builtin_amdgcn_wmma_f32_16x16x4_f32(
          false, a[k], false, b, (short)0, sacc, false, false);
    }

    // ---- mask + online softmax bookkeeping (C layout: row = half*8+r, col = n) ----
    int kpos = kt * 16 + n;
    float p[8], rs[8], sc[8], mn_[8];
    #pragma unroll
    for (int r = 0; r < 8; ++r) {
      int rr  = half * 8 + r;
      float s = (kpos <= lqi[rr]) ? sacc[r] * 0.125f : -INFINITY;  // 1/sqrt(64)
      float tm = s;
      #pragma unroll
      for (int off = 8; off > 0; off >>= 1) tm = fmaxf(tm, __shfl_xor(tm, off, 32));
      float mold = lm[rr];
      float mn   = fmaxf(mold, tm);
      float pe   = expf(s - mn);
      float sum  = pe;
      #pragma unroll
      for (int off = 8; off > 0; off >>= 1) sum += __shfl_xor(sum, off, 32);
      p[r] = pe; rs[r] = sum; mn_[r] = mn; sc[r] = expf(mold - mn);
    }

    __syncthreads();   // prior iteration's readers of ptile/lscale are done
    #pragma unroll
    for (int r = 0; r < 8; ++r) {
      int rr = half * 8 + r;
      ptile[rr * 16 + n] = p[r];
      if (n == 0) {
        lm[rr]     = mn_[r];
        lscale[rr] = sc[r];
        ll[rr]     = ll[rr] * sc[r] + rs[r];
      }
    }
    __syncthreads();   // publish ptile / lscale

    // ---- accumulate P @ V; each lane owns output dims {2*lane, 2*lane+1} ----
    #pragma unroll
    for (int r = 0; r < 16; ++r) {
      float f = lscale[r];
      oacc[r].x *= f; oacc[r].y *= f;
    }
    for (int j = 0; j < 16; ++j) {
      const float2 v = *reinterpret_cast<const float2*>(
          valb + (size_t)(kt * 16 + j) * D_ + 2 * lane);
      #pragma unroll
      for (int r = 0; r < 16; ++r) {
        float pv = ptile[r * 16 + j];
        oacc[r].x += pv * v.x;
        oacc[r].y += pv * v.y;
      }
    }
  }

  __syncthreads();
  #pragma unroll
  for (int r = 0; r < 16; ++r) {
    int grow = rt * 16 + r;
    if (grow < NQ_) {
      float inv = 1.0f / ll[r];
      float* op = out + ((size_t)bh * L_ + lqi[r]) * D_ + 2 * lane;
      op[0] = oacc[r].x * inv;
      op[1] = oacc[r].y * inv;
    }
  }
}

// ---------------------------------------------------------------------------
extern "C" void kernel_launch(void* const* d_in, const int* in_sizes, int n_in,
                              void* d_out, int out_size, void* d_ws, size_t ws_size,
                              hipStream_t stream) {
  const float* query     = (const float*)d_in[0];
  const float* key       = (const float*)d_in[1];
  const float* value     = (const float*)d_in[2];
  const int*   key_index = (const int*)d_in[3];
  float*       out       = (float*)d_out;

  int nk = in_sizes[3] / L_;                      // = 45

  float* ws_sp = (float*)d_ws;                    // B*H*L floats (256 KB)
  int*   ws_qi = (int*)(ws_sp + (size_t)B_ * H_ * L_);  // B*H*45 ints

  sparsity_kernel<<<(B_ * H_ * L_) / 256, 256, 0, stream>>>(query, key, key_index, ws_sp, nk);
  topk_kernel<<<B_ * H_, 256, 0, stream>>>(ws_sp, ws_qi);
  cumsum_kernel<<<B_ * H_, D_, 0, stream>>>(value, out);
  attn_kernel<<<B_ * H_ * 3, 32, 0, stream>>>(query, key, value, ws_qi, out);
}